// LCNSpiking_58162447123130
// MI455X (gfx1250) — compile-verified
//
#include <hip/hip_runtime.h>

#define T_STEPS 20
#define IN_DIM  14400
#define KNN     25
#define BLK     512
#define HPAD    464            // 450 padded to a multiple of 4 (16B-aligned rows)
#define FCW     4              // waves in the FC kernel

typedef float v2f __attribute__((ext_vector_type(2)));
typedef float v8f __attribute__((ext_vector_type(8)));

// One gather-weight-reduce layer: hin (LDS) -> hout (LDS).
__device__ __forceinline__ void lcn_layer(const float* __restrict__ hin,
                                          float* __restrict__ hout,
                                          const float* __restrict__ w,
                                          const float* __restrict__ b,
                                          const int* __restrict__ knn,
                                          int dout)
{
    for (int j = (int)threadIdx.x; j < dout; j += BLK) {
        const float* wj = w   + (size_t)j * KNN;
        const int*   kj = knn + (size_t)j * KNN;
        float acc = 0.0f;
#pragma unroll
        for (int k = 0; k < KNN; ++k) {
            acc = fmaf(hin[kj[k]], wj[k], acc);   // ds_load gather + v_fmac
        }
        hout[j] = acc + b[j];
    }
}

// One workgroup per surviving row (b, t = T-1). All activations staged in LDS.
__global__ __launch_bounds__(BLK) void lcn_fused(
    const float* __restrict__ x,
    const float* __restrict__ w0, const float* __restrict__ b0, const int* __restrict__ k0,
    const float* __restrict__ w1, const float* __restrict__ b1, const int* __restrict__ k1,
    const float* __restrict__ w2, const float* __restrict__ b2, const int* __restrict__ k2,
    const float* __restrict__ w3, const float* __restrict__ b3, const int* __restrict__ k3,
    const float* __restrict__ w4, const float* __restrict__ b4, const int* __restrict__ k4,
    float* __restrict__ H)                         // [32, HPAD] scratch (pad zero-filled)
{
    __shared__ float bufX[IN_DIM];                 // 57.6 KB
    __shared__ float bufA[7200];                   // 28.8 KB  (86.4 KB total < 320 KB WGP LDS)

    const int row = (int)blockIdx.x;               // 0..31 == batch index
    const float* src = x + ((size_t)row * T_STEPS + (T_STEPS - 1)) * IN_DIM;

    // --- Stage input row into LDS via CDNA5 async global->LDS DMA (ASYNCcnt) ---
    unsigned ldsBase = (unsigned)(size_t)&bufX[0]; // low 32 bits of generic ptr == LDS addr
    for (int c = (int)threadIdx.x; c < IN_DIM / 4; c += BLK) {
        unsigned laddr = ldsBase + (unsigned)c * 16u;
        const float* g = src + (size_t)c * 4;
        asm volatile("global_load_async_to_lds_b128 %0, %1, off"
                     :: "v"(laddr), "v"(g)
                     : "memory");
    }
    asm volatile("s_wait_asynccnt 0x0" ::: "memory");
    __syncthreads();

    lcn_layer(bufX, bufA, w0, b0, k0, 7200);  __syncthreads();   // 14400 -> 7200
    lcn_layer(bufA, bufX, w1, b1, k1, 3600);  __syncthreads();   //  7200 -> 3600
    lcn_layer(bufX, bufA, w2, b2, k2, 1800);  __syncthreads();   //  3600 -> 1800
    lcn_layer(bufA, bufX, w3, b3, k3,  900);  __syncthreads();   //  1800 ->  900
    lcn_layer(bufX, bufA, w4, b4, k4,  450);  __syncthreads();   //   900 ->  450

    // padded store: WMMA kernel reads a branch-free, zero-padded [32, HPAD] matrix
    for (int j = (int)threadIdx.x; j < HPAD; j += BLK)
        H[(size_t)row * HPAD + j] = (j < 450) ? bufA[j] : 0.0f;
}

// Dense FC tail on the matrix pipe: out[32,2] = H[32,450] @ Wfc[450,2] + bfc.
// 4 waves split K; V_WMMA_F32_16X16X4_F32 on LDS-resident, zero-padded operands.
__global__ __launch_bounds__(32 * FCW) void fc_wmma(
    const float* __restrict__ H,      // [32, HPAD], pad already zero
    const float* __restrict__ Wfc,    // [450, 2]
    const float* __restrict__ bfc,    // [2]
    float* __restrict__ out)          // [32, 2]
{
    __shared__ float Hl[32 * HPAD];           // 59.4 KB
    __shared__ float Bl[HPAD * 16];           // 29.7 KB (dense padded B, zeros for N>=2)
    __shared__ float Rbuf[FCW * 2 * 8 * 32];  // 8 KB    (per-wave partial D tiles)

    const int tid  = (int)threadIdx.x;
    const int lane = tid & 31;
    const int wave = tid >> 5;

    // async-stage H into LDS (16B chunks, ASYNCcnt)
    {
        unsigned ldsBase = (unsigned)(size_t)&Hl[0];
        for (int c = tid; c < (32 * HPAD) / 4; c += 32 * FCW) {
            unsigned laddr = ldsBase + (unsigned)c * 16u;
            const float* g = H + (size_t)c * 4;
            asm volatile("global_load_async_to_lds_b128 %0, %1, off"
                         :: "v"(laddr), "v"(g)
                         : "memory");
        }
        asm volatile("s_wait_asynccnt 0x0" ::: "memory");
    }
    // build dense padded B: Bl[k*16+n] = (n<2 && k<450) ? Wfc[k*2+n] : 0
    for (int i = tid; i < HPAD * 16; i += 32 * FCW) {
        const int k = i >> 4, n = i & 15;
        Bl[i] = (n < 2 && k < 450) ? Wfc[(size_t)k * 2 + n] : 0.0f;
    }
    __syncthreads();

    const int mn    = lane & 15;      // M for A / N for B,D
    const int half  = lane >> 4;
    const int khalf = half * 2;
    const float* A0 = &Hl[(size_t)mn * HPAD];
    const float* A1 = &Hl[(size_t)(mn + 16) * HPAD];

    v8f acc0 = {};
    v8f acc1 = {};
    const int kBeg = wave * (HPAD / FCW);           // 116 K per wave, %4 == 0
    const int kEnd = kBeg + (HPAD / FCW);
    for (int kb0 = kBeg; kb0 < kEnd; kb0 += 4) {
        const int ka = kb0 + khalf;
        v2f a0, a1, bmat;
        a0.x = A0[ka];  a0.y = A0[ka + 1];          // ds_load_b64
        a1.x = A1[ka];  a1.y = A1[ka + 1];
        bmat.x = Bl[ka * 16 + mn];
        bmat.y = Bl[(ka + 1) * 16 + mn];
        acc0 = __builtin_amdgcn_wmma_f32_16x16x4_f32(false, a0, false, bmat,
                                                     (short)0, acc0, false, false);
        acc1 = __builtin_amdgcn_wmma_f32_16x16x4_f32(false, a1, false, bmat,
                                                     (short)0, acc1, false, false);
    }

    // stash per-wave partial D tiles, then wave 0 reduces and stores
    float* r0 = &Rbuf[(size_t)(wave * 2 + 0) * 8 * 32];
    float* r1 = &Rbuf[(size_t)(wave * 2 + 1) * 8 * 32];
#pragma unroll
    for (int r = 0; r < 8; ++r) {
        r0[r * 32 + lane] = acc0[r];
        r1[r * 32 + lane] = acc1[r];
    }
    __syncthreads();

    if (wave == 0 && mn < 2) {
        // D layout: VGPR r -> M=r (lanes 0-15) / M=r+8 (lanes 16-31), N = lane%16
        const int n     = mn;
        const int mbase = half * 8;
        const float bias = bfc[n];
#pragma unroll
        for (int r = 0; r < 8; ++r) {
            float s0 = 0.0f, s1 = 0.0f;
#pragma unroll
            for (int w = 0; w < FCW; ++w) {
                s0 += Rbuf[(size_t)((w * 2 + 0) * 8 + r) * 32 + lane];
                s1 += Rbuf[(size_t)((w * 2 + 1) * 8 + r) * 32 + lane];
            }
            out[(size_t)(mbase + r) * 2 + n]      = s0 + bias;
            out[(size_t)(16 + mbase + r) * 2 + n] = s1 + bias;
        }
    }
}

extern "C" void kernel_launch(void* const* d_in, const int* in_sizes, int n_in,
                              void* d_out, int out_size, void* d_ws, size_t ws_size,
                              hipStream_t stream) {
    (void)in_sizes; (void)n_in; (void)out_size; (void)ws_size;

    const float* x    = (const float*)d_in[0];
    const float* w0   = (const float*)d_in[1];
    const float* b0   = (const float*)d_in[2];
    const int*   knn0 = (const int*)  d_in[3];
    const float* w1   = (const float*)d_in[4];
    const float* b1   = (const float*)d_in[5];
    const int*   knn1 = (const int*)  d_in[6];
    const float* w2   = (const float*)d_in[7];
    const float* b2   = (const float*)d_in[8];
    const int*   knn2 = (const int*)  d_in[9];
    const float* w3   = (const float*)d_in[10];
    const float* b3   = (const float*)d_in[11];
    const int*   knn3 = (const int*)  d_in[12];
    const float* w4   = (const float*)d_in[13];
    const float* b4   = (const float*)d_in[14];
    const int*   knn4 = (const int*)  d_in[15];
    const float* Wfc  = (const float*)d_in[16];
    const float* bfc  = (const float*)d_in[17];

    float* H = (float*)d_ws;            // 32*HPAD floats = 59.4 KB scratch

    lcn_fused<<<32, BLK, 0, stream>>>(x,
                                      w0, b0, knn0,
                                      w1, b1, knn1,
                                      w2, b2, knn2,
                                      w3, b3, knn3,
                                      w4, b4, knn4,
                                      H);
    fc_wmma<<<1, 32 * FCW, 0, stream>>>(H, Wfc, bfc, (float*)d_out);
}